// AliasFreeSampling_85392539779309
// MI455X (gfx1250) — compile-verified
//
#include <hip/hip_runtime.h>

// ---------------------------------------------------------------------------
// Alias-free downsample: separable 65-tap FIR (reflect pad 32) + 2x2 avg pool,
// fused into two stride-2 66-tap passes, each mapped onto the CDNA5 fp32
// matrix pipe as banded GEMMs via V_WMMA_F32_16X16X4_F32.
// Tile staging uses GLOBAL_LOAD_ASYNC_TO_LDS_B32 (ASYNCcnt-tracked) so the
// reflect-mapped gather lands directly in LDS.
// ---------------------------------------------------------------------------

typedef float v2f __attribute__((ext_vector_type(2)));
typedef float v8f __attribute__((ext_vector_type(8)));

namespace {
constexpr int Hdim = 512;
constexpr int Wdim = 512;
constexpr int OH   = 256;
constexpr int OW   = 256;
constexpr int NPLANE = 8 * 32;          // B*C

// Workgroup output tile: 32 rows x 64 cols (of the 256x256 output plane).
constexpr int TO_R = 32;                // out rows per WG
constexpr int TO_C = 64;                // out cols per WG
// Input tile (full res, incl. 66-tap stride-2 halo): rows = 2*32-2+66 = 128,
// cols = 2*64-2+66 = 192.
constexpr int XT_R = 128;
constexpr int XT_C = 192;
constexpr int XP   = XT_C + 4;          // 196: row stride mod 64 == 4 -> b64 A loads conflict-free
// Horizontal-pass intermediate: 128 rows x 64 cols.
constexpr int TT_R = 128;
constexpr int TT_C = 64;
constexpr int TP   = 76;                // mod 64 == 12: D-stores (rows +8) and B-loads (rows +2) bank-disjoint

constexpr int KSTEPS = 24;              // 96-wide band / K=4 per WMMA
}

__device__ __forceinline__ int reflect512(int v) {
  // numpy 'reflect' (no edge repeat), pad <= 32 so one fold suffices
  v = (v < 0) ? -v : v;
  v = (v >= Hdim) ? (2 * Hdim - 2 - v) : v;
  return v;
}

__global__ __launch_bounds__(256)
void alias_free_down_kernel(const float* __restrict__ x,
                            const float* __restrict__ kern,   // 65 taps
                            float* __restrict__ out) {
  __shared__ float xs[XT_R * XP];   // 100352 B
  __shared__ float ts[TT_R * TP];   //  38912 B  (total 136 KB LDS)

  const int plane  = blockIdx.x >> 5;       // 0..255
  const int t5     = blockIdx.x & 31;
  const int tile_i = t5 >> 2;               // 0..7
  const int tile_j = t5 & 3;                // 0..3
  const int i0 = tile_i * TO_R;             // out-row origin
  const int j0 = tile_j * TO_C;             // out-col origin

  const int tid  = threadIdx.x;
  const int lane = tid & 31;
  const int wave = tid >> 5;
  const int mn   = lane & 15;               // M (A) / N (B,C,D) index
  const int hsel = (lane >> 4) & 1;         // half-wave select (K offset +2)

  // -------------------------------------------------------------------------
  // Per-lane band-matrix registers. Merged stride-2 kernel:
  //   k2[s] = 0.5*(k[s] + k[s-1]),  s = 0..65,  k[-1]=k[65]=0
  // Band value at (band col c', lane index mn): k2[c' - 2*mn].
  // Identical register image serves as B (horizontal) and A (vertical).
  // -------------------------------------------------------------------------
  float kbv[KSTEPS][2];
#pragma unroll
  for (int kt = 0; kt < KSTEPS; ++kt) {
#pragma unroll
    for (int v = 0; v < 2; ++v) {
      const int c = 4 * kt + 2 * hsel + v;
      const int s = c - 2 * mn;
      float val = 0.0f;
      if (s >= 0 && s <= 65) {
        const float a = (s <= 64) ? kern[s] : 0.0f;
        const float b = (s >= 1) ? kern[s - 1] : 0.0f;
        val = 0.5f * (a + b);
      }
      kbv[kt][v] = val;
    }
  }

  // -------------------------------------------------------------------------
  // Stage 1: stage 128 x 192 reflect-mapped input tile straight into LDS via
  // async-to-LDS loads (per-lane global addr + per-lane LDS addr; ASYNCcnt).
  // local (r, c) <-> global (2*i0 - 32 + r, 2*j0 - 32 + c)
  // 24576 elements == 96 * 256 threads -> uniform trip count, EXEC all-ones.
  // -------------------------------------------------------------------------
  const float* __restrict__ xin = x + (size_t)plane * (Hdim * Wdim);
  for (int idx = tid; idx < XT_R * XT_C; idx += 256) {
    const int r = idx / XT_C;
    const int c = idx - r * XT_C;
    const int gr = reflect512(2 * i0 - 32 + r);
    const int gc = reflect512(2 * j0 - 32 + c);
    const unsigned int       ldsa = (unsigned int)(size_t)(&xs[r * XP + c]);
    const unsigned long long ga   = (unsigned long long)(size_t)(xin + gr * Wdim + gc);
    asm volatile("global_load_async_to_lds_b32 %0, %1, off"
                 :: "v"(ldsa), "v"(ga) : "memory");
  }
  asm volatile("s_wait_asynccnt 0x0" ::: "memory");
  __syncthreads();

  // -------------------------------------------------------------------------
  // Stage 2: horizontal stride-2 FIR as banded GEMM.
  //   t[r, 16*jb + n] = sum_{c'} xs[r, 32*jb + c'] * k2[c' - 2n]
  // 8 r-blocks x 4 j-blocks = 32 WMMA tiles; constant 4-trip loop per wave so
  // control flow stays wave-uniform (WMMA requires EXEC == all-ones).
  // -------------------------------------------------------------------------
#pragma unroll 1
  for (int q = 0; q < 4; ++q) {
    const int t  = (q << 3) + wave;
    const int rb = t >> 2;
    const int jb = t & 3;
    const int r0 = rb << 4;
    const int cb = jb << 5;                 // 32*jb
    const int abase = (r0 + mn) * XP + cb + 2 * hsel;

    v8f acc = {};
#pragma unroll
    for (int kt = 0; kt < KSTEPS; ++kt) {
      const v2f a = *(const v2f*)(&xs[abase + 4 * kt]);    // ds_load b64
      const v2f b = {kbv[kt][0], kbv[kt][1]};
      acc = __builtin_amdgcn_wmma_f32_16x16x4_f32(
          false, a, false, b, (short)0, acc, false, false);
    }

    const int scol  = (jb << 4) + mn;
    const int srow0 = r0 + (hsel << 3);
#pragma unroll
    for (int v = 0; v < 8; ++v) {
      ts[(srow0 + v) * TP + scol] = acc[v];
    }
  }
  __syncthreads();

  // -------------------------------------------------------------------------
  // Stage 3: vertical stride-2 FIR as banded GEMM (band matrix now operand A).
  //   out[16*ib + m, 16*jb2 + n] = sum_{c'} k2[c' - 2m] * ts[32*ib + c', 16*jb2 + n]
  // 2 x 4 = 8 WMMA tiles, one per wave.
  // -------------------------------------------------------------------------
  {
    const int ib  = wave >> 2;              // 0..1
    const int jb2 = wave & 3;               // 0..3
    const int tb  = ib << 5;                // 32*ib
    const int ncol = (jb2 << 4) + mn;

    v8f acc = {};
#pragma unroll
    for (int kt = 0; kt < KSTEPS; ++kt) {
      const int crow = tb + 4 * kt + 2 * hsel;
      const v2f a = {kbv[kt][0], kbv[kt][1]};
      v2f b;
      b.x = ts[crow * TP + ncol];
      b.y = ts[(crow + 1) * TP + ncol];
      acc = __builtin_amdgcn_wmma_f32_16x16x4_f32(
          false, a, false, b, (short)0, acc, false, false);
    }

    const int orow0 = i0 + (ib << 4) + (hsel << 3);
    const int ocol  = j0 + (jb2 << 4) + mn;
    float* __restrict__ op = out + (size_t)plane * (OH * OW) + ocol;
#pragma unroll
    for (int v = 0; v < 8; ++v) {
      op[(size_t)(orow0 + v) * OW] = acc[v];
    }
  }
}

extern "C" void kernel_launch(void* const* d_in, const int* in_sizes, int n_in,
                              void* d_out, int out_size, void* d_ws, size_t ws_size,
                              hipStream_t stream) {
  (void)in_sizes; (void)n_in; (void)out_size; (void)d_ws; (void)ws_size;
  const float* x    = (const float*)d_in[0];
  const float* kern = (const float*)d_in[1];
  float* out        = (float*)d_out;

  // 256 planes * (8 row-tiles x 4 col-tiles) = 8192 workgroups, 256 threads.
  dim3 grid(NPLANE * 32);
  dim3 block(256);
  alias_free_down_kernel<<<grid, block, 0, stream>>>(x, kern, out);
}